// GNN_75118978007185
// MI455X (gfx1250) — compile-verified
//
#include <hip/hip_runtime.h>

// ---------------------------------------------------------------------------
// GNN graph-reasoning forward for MI455X (gfx1250, wave32).
// 14 tiny fp32 GEMMs (M=10, K in {20,30,50}, N=131072) + BN + ReLU, some
// summed pairwise. Memory-bound (~240 MB @ 23.3 TB/s ~ 10us). Math via
// V_WMMA_F32_16X16X4_F32 (full fp32 precision, 1024 MACs/instr).
//
// Round-2 fix: all global loads are UNCONDITIONAL (clamped addresses), with
// zeroing done only in the A fragment via branch-free selects. Tail-K rows
// contribute nothing because their A columns are zero, so B may load any
// valid address. Conv identity is a template parameter, selected by a
// uniform switch on blockIdx.z -> no divergent control flow anywhere in the
// inner loop, EXEC stays all-ones for every WMMA.
// ---------------------------------------------------------------------------

typedef float v2f __attribute__((ext_vector_type(2)));
typedef float v8f __attribute__((ext_vector_type(8)));

#define HW   16384   // H*W
#define CB   10      // channels per node
#define NB   8       // batch
constexpr int TILES = 4;   // 16-pixel N-tiles per wave -> 64 pixels/wave

struct Params {
  const float* p;        // p_nodes [7,B,C,H,W]
  const float* h;        // h_nodes [3,B,C,H,W]
  const float* f;        // f_nodes [2,B,C,H,W]
  const float* W[8];     // weights, conv_specs order
  const float* bn[8];    // bn stats [4,C], conv_specs order
  float*       out;      // (p_new 7 | h_new 3 | f_new 2) * [B,C,H,W]
};

// spec order: 0=comp_full 1=bg_f 2=decomp_full 3=comp_u 4=comp_l 5=bg_h 6=decomp_half 7=bg_p
constexpr int kConvK[14] = {20,30,30,50,20,30,20,20,20,20,20,20,20,20};
constexpr int kConvW[14] = {1,0,5,3,2,4,2,7,6,6,6,6,6,6};

// (conv, input-channel k) -> base pointer of that channel's [H,W] plane.
// CONV is compile-time; only cheap per-lane selects remain at runtime.
template<int CONV>
__device__ __forceinline__ const float* row_ptr(const Params& P, int k, int b) {
  int t, n, c;
  if constexpr (CONV == 0)      { if (k < 10) { t=2; n=0; c=k; } else { t=1; n=0; c=k-10; } }
  else if constexpr (CONV == 1) { if (k < 10) { t=2; n=1; c=k; }
                                  else if (k < 20) { t=1; n=1; c=k-10; }
                                  else { t=1; n=2; c=k-20; } }
  else if constexpr (CONV == 2) { if (k < 10) { t=2; n=0; c=k; }
                                  else if (k < 20) { t=1; n=0; c=k-10; }
                                  else { t=0; n=0; c=k-20; } }
  else if constexpr (CONV == 3) { if (k < 40) { t=0; n=1+(k&3); c=k>>2; } else { t=1; n=1; c=k-40; } }
  else if constexpr (CONV == 4) { if (k < 10) { t=2; n=1; c=k; } else { t=1; n=1; c=k-10; } }
  else if constexpr (CONV == 5) { if (k < 20) { t=0; n=5+(k&1); c=k>>1; } else { t=1; n=2; c=k-20; } }
  else if constexpr (CONV == 6) { if (k < 10) { t=2; n=1; c=k; } else { t=1; n=2; c=k-10; } }
  else if constexpr (CONV == 7) { if (k < 10) { t=1; n=0; c=k; } else { t=0; n=0; c=k-10; } }
  else if constexpr (CONV >= 12){ if (k < 10) { t=1; n=2; c=k; } else { t=0; n=CONV-7; c=k-10; } }
  else                          { if (k < 10) { t=1; n=1; c=k; } else { t=0; n=CONV-7; c=k-10; } }
  const float* base = (t == 0) ? P.p : ((t == 1) ? P.h : P.f);
  return base + (((n * NB + b) * CB + c) * HW);
}

template<int CONV>
__device__ __forceinline__ void run_conv(const Params& P, int b, int pix0,
                                         int lh /*lane>=16*/, int M /*lane&15*/,
                                         float res[TILES][8]) {
  constexpr int K  = kConvK[CONV];
  constexpr int NK = (K + 3) / 4;
  const float* __restrict__ Wm  = P.W[kConvW[CONV]];
  const float* __restrict__ bnp = P.bn[kConvW[CONV]];

  v8f acc[TILES];
  #pragma unroll
  for (int t = 0; t < TILES; ++t) acc[t] = v8f{0.f,0.f,0.f,0.f,0.f,0.f,0.f,0.f};

  const int Mw = (M < CB) ? M : (CB - 1);          // clamped weight row

  for (int kc = 0; kc < NK; ++kc) {
    // A 16x4 fragment: lanes0-15 hold K{0,1}, lanes16-31 K{2,3}.
    // B 4x16 fragment: vgpr v holds K = v + 2*half  -> same k0,k1 per lane.
    const int k0  = kc * 4 + 2 * lh;
    const int k1  = k0 + 1;
    const int k0c = (k0 < K) ? k0 : (K - 1);       // clamp for address only
    const int k1c = (k1 < K) ? k1 : (K - 1);

    // Unconditional (clamped) loads; zeroing by branch-free select in A only.
    float a0 = Wm[Mw * K + k0c];
    float a1 = Wm[Mw * K + k1c];
    a0 = (M < CB && k0 < K) ? a0 : 0.f;
    a1 = (M < CB && k1 < K) ? a1 : 0.f;
    v2f A; A.x = a0; A.y = a1;

    const float* r0 = row_ptr<CONV>(P, k0c, b);
    const float* r1 = row_ptr<CONV>(P, k1c, b);

    #pragma unroll
    for (int t = 0; t < TILES; ++t) {
      const int pix = pix0 + t * 16 + M;           // N index within tile
      v2f Bv;
      Bv.x = r0[pix];                              // unconditional loads
      Bv.y = r1[pix];
      acc[t] = __builtin_amdgcn_wmma_f32_16x16x4_f32(
          false, A, false, Bv, (short)0, acc[t], false, false);
    }
  }

  // BN scale/shift per output channel o = v + 8*half (C/D layout), then ReLU.
  float sc[8], sh[8];
  #pragma unroll
  for (int v = 0; v < 8; ++v) {
    const int o  = v + 8 * lh;
    const int oc = (o < CB) ? o : (CB - 1);        // clamped, unconditional loads
    const float g  = bnp[oc];
    const float be = bnp[CB + oc];
    const float mu = bnp[2 * CB + oc];
    const float va = bnp[3 * CB + oc];
    const float s  = g * rsqrtf(va + 1e-5f);
    sc[v] = (o < CB) ? s : 0.f;
    sh[v] = (o < CB) ? (be - mu * s) : 0.f;
  }
  #pragma unroll
  for (int t = 0; t < TILES; ++t)
    #pragma unroll
    for (int v = 0; v < 8; ++v)
      res[t][v] += fmaxf(acc[t][v] * sc[v] + sh[v], 0.f);
}

template<int CA, int CBv, int G>
__device__ __forceinline__ void run_job(const Params& P, int b, int pix0,
                                        int lh, int M) {
  float res[TILES][8];
  #pragma unroll
  for (int t = 0; t < TILES; ++t)
    #pragma unroll
    for (int v = 0; v < 8; ++v) res[t][v] = 0.f;

  run_conv<CA>(P, b, pix0, lh, M, res);
  if constexpr (CBv >= 0) run_conv<CBv>(P, b, pix0, lh, M, res);  // h1/h2 sums

  float* __restrict__ out = P.out + ((size_t)(G * NB + b) * CB) * HW;
  #pragma unroll
  for (int v = 0; v < 8; ++v) {
    const int o = v + 8 * lh;
    if (o < CB) {                                   // half-wave predicated store
      #pragma unroll
      for (int t = 0; t < TILES; ++t)
        out[o * HW + pix0 + t * 16 + M] = res[t][v];
    }
  }
}

__global__ __launch_bounds__(256)
void GNN_75118978007185_kernel(Params P) {
  const int lane = threadIdx.x & 31;
  const int wave = threadIdx.x >> 5;
  const int lh   = lane >> 4;
  const int M    = lane & 15;
  const int b    = blockIdx.y;
  const int pix0 = (blockIdx.x * 8 + wave) * (TILES * 16);

  switch (blockIdx.z) {                             // uniform per block
    case 0:  run_job<0,  -1, 10>(P, b, pix0, lh, M); break;  // f0
    case 1:  run_job<1,  -1, 11>(P, b, pix0, lh, M); break;  // f1
    case 2:  run_job<2,  -1,  7>(P, b, pix0, lh, M); break;  // h0
    case 3:  run_job<3,   4,  8>(P, b, pix0, lh, M); break;  // h1 = comp_u + decomp_full
    case 4:  run_job<5,   6,  9>(P, b, pix0, lh, M); break;  // h2 = comp_l + decomp_full
    case 5:  run_job<7,  -1,  0>(P, b, pix0, lh, M); break;  // p0
    case 6:  run_job<8,  -1,  1>(P, b, pix0, lh, M); break;  // p1
    case 7:  run_job<9,  -1,  2>(P, b, pix0, lh, M); break;  // p2
    case 8:  run_job<10, -1,  3>(P, b, pix0, lh, M); break;  // p3
    case 9:  run_job<11, -1,  4>(P, b, pix0, lh, M); break;  // p4
    case 10: run_job<12, -1,  5>(P, b, pix0, lh, M); break;  // p5
    case 11: run_job<13, -1,  6>(P, b, pix0, lh, M); break;  // p6
  }
}

extern "C" void kernel_launch(void* const* d_in, const int* in_sizes, int n_in,
                              void* d_out, int out_size, void* d_ws, size_t ws_size,
                              hipStream_t stream) {
  (void)in_sizes; (void)n_in; (void)out_size; (void)d_ws; (void)ws_size;
  Params P;
  P.p = (const float*)d_in[0];
  P.h = (const float*)d_in[1];
  P.f = (const float*)d_in[2];
  // d_in[3..5] = xp/xh/xf (unused by the reference graph)
  const int wslot[8] = {6, 8, 10, 12, 14, 16, 18, 20};  // spec-order W indices
  for (int i = 0; i < 8; ++i) {
    P.W[i]  = (const float*)d_in[wslot[i]];
    P.bn[i] = (const float*)d_in[wslot[i] + 1];
  }
  P.out = (float*)d_out;

  dim3 grid(HW / (8 * TILES * 16), NB, 12);   // (32 pixel blocks, 8 batches, 12 jobs)
  dim3 block(256);                             // 8 wave32s
  GNN_75118978007185_kernel<<<grid, block, 0, stream>>>(P);
}